// ThreeLayerGAT_BN_75977971466897
// MI455X (gfx1250) — compile-verified
//
#include <hip/hip_runtime.h>
#include <math.h>
#include <float.h>

#define NN 50000
#define EE 800000

typedef __attribute__((ext_vector_type(2))) float v2f;
typedef __attribute__((ext_vector_type(8))) float v8f;

// ---------------------------------------------------------------------------
// GEMM: H[N,128] = X[N,K] @ W[K,128], fp32 via V_WMMA_F32_16X16X4_F32.
// One wave handles a 16-row stripe and all 8 column tiles (16 cols each).
// A frag (16x4 f32): lanes 0-15 rows M=0..15 holding K={0,1}; lanes 16-31
// same rows holding K={2,3}. B frag symmetric over N columns. C/D: VGPR j =
// row j (lanes 0-15) / row j+8 (lanes 16-31).
// ---------------------------------------------------------------------------
__global__ __launch_bounds__(128) void gemm_wmma_f32(
    const float* __restrict__ X, const float* __restrict__ W,
    float* __restrict__ Hout, int K) {
  int gid  = blockIdx.x * blockDim.x + threadIdx.x;
  int wave = gid >> 5;
  int lane = threadIdx.x & 31;
  int row0 = wave * 16;
  if (row0 >= NN) return;               // uniform per-wave exit, EXEC stays full

  int half = lane >> 4;                 // 0: K pair {0,1}, 1: K pair {2,3}
  int l16  = lane & 15;

  v8f acc[8];
#pragma unroll
  for (int t = 0; t < 8; ++t) acc[t] = (v8f){0.f,0.f,0.f,0.f,0.f,0.f,0.f,0.f};

  const float* xrow = X + (size_t)(row0 + l16) * K;

  for (int k = 0; k < K; k += 4) {
    v2f a;
    a.x = xrow[k + 2 * half];
    a.y = xrow[k + 2 * half + 1];
#pragma unroll
    for (int t = 0; t < 8; ++t) {
      const float* wp = W + (size_t)(k + 2 * half) * 128 + t * 16 + l16;
      v2f b;
      b.x = wp[0];
      b.y = wp[128];
      acc[t] = __builtin_amdgcn_wmma_f32_16x16x4_f32(
          false, a, false, b, (short)0, acc[t], false, false);
    }
  }

#pragma unroll
  for (int t = 0; t < 8; ++t) {
#pragma unroll
    for (int j = 0; j < 8; ++j) {
      int r = row0 + j + 8 * half;
      Hout[(size_t)r * 128 + t * 16 + l16] = acc[t][j];
    }
  }
}

// ---------------------------------------------------------------------------
// el/er: per (node, head) attention logit halves.
// ---------------------------------------------------------------------------
__global__ void eler_kernel(const float* __restrict__ h,
                            const float* __restrict__ al,
                            const float* __restrict__ ar,
                            float* __restrict__ el, float* __restrict__ er) {
  int t = blockIdx.x * blockDim.x + threadIdx.x;
  if (t >= NN * 2) return;
  int n = t >> 1, hd = t & 1;
  const float* hp  = h + (size_t)n * 128 + hd * 64;
  const float* alp = al + hd * 64;
  const float* arp = ar + hd * 64;
  float sl = 0.f, sr = 0.f;
#pragma unroll 4
  for (int f = 0; f < 64; ++f) {
    float v = hp[f];
    sl += v * alp[f];
    sr += v * arp[f];
  }
  el[t] = sl;
  er[t] = sr;
}

// ---------------------------------------------------------------------------
// init: agg = 0 over N*128; m = -FLT_MAX, s = 0 over N*2.
// ---------------------------------------------------------------------------
__global__ void init_kernel(float* __restrict__ agg, float* __restrict__ m,
                            float* __restrict__ s) {
  int t = blockIdx.x * blockDim.x + threadIdx.x;
  if (t < NN * 128) agg[t] = 0.f;
  if (t < NN * 2) { m[t] = -FLT_MAX; s[t] = 0.f; }
}

__device__ __forceinline__ void atomicMaxF(float* addr, float val) {
  if (val >= 0.f) atomicMax((int*)addr, __float_as_int(val));
  else            atomicMin((unsigned int*)addr, __float_as_uint(val));
}

// ---------------------------------------------------------------------------
// Edge pass 1: e = leaky_relu(el[src]+er[dst]); m[dst] = max(m[dst], e).
// ---------------------------------------------------------------------------
__global__ void edge_pass1(const int* __restrict__ src, const int* __restrict__ dst,
                           const float* __restrict__ el, const float* __restrict__ er,
                           float* __restrict__ ebuf, float* __restrict__ mbuf) {
  int i = blockIdx.x * blockDim.x + threadIdx.x;
  if (i >= EE) return;
  int s = src[i], d = dst[i];
  float e0 = el[2 * s]     + er[2 * d];
  float e1 = el[2 * s + 1] + er[2 * d + 1];
  e0 = e0 > 0.f ? e0 : 0.2f * e0;
  e1 = e1 > 0.f ? e1 : 0.2f * e1;
  ebuf[2 * i]     = e0;
  ebuf[2 * i + 1] = e1;
  atomicMaxF(mbuf + 2 * d,     e0);
  atomicMaxF(mbuf + 2 * d + 1, e1);
}

// ---------------------------------------------------------------------------
// Edge pass 2: w = exp(e - m[dst]); s[dst] += w; ebuf overwritten with w.
// ---------------------------------------------------------------------------
__global__ void edge_pass2(const int* __restrict__ dst, float* __restrict__ ebuf,
                           const float* __restrict__ mbuf, float* __restrict__ sbuf) {
  int i = blockIdx.x * blockDim.x + threadIdx.x;
  if (i >= EE) return;
  int d = dst[i];
  float w0 = __expf(ebuf[2 * i]     - mbuf[2 * d]);
  float w1 = __expf(ebuf[2 * i + 1] - mbuf[2 * d + 1]);
  ebuf[2 * i]     = w0;
  ebuf[2 * i + 1] = w1;
  atomicAdd(sbuf + 2 * d,     w0);
  atomicAdd(sbuf + 2 * d + 1, w1);
}

// ---------------------------------------------------------------------------
// Edge pass 3: one wave per edge; lane handles 4 channels.
// agg[dst] += h[src] * (w / s[dst]) per head.
// ---------------------------------------------------------------------------
__global__ void edge_pass3(const int* __restrict__ src, const int* __restrict__ dst,
                           const float* __restrict__ ebuf, const float* __restrict__ sbuf,
                           const float* __restrict__ h, float* __restrict__ agg) {
  int gid  = blockIdx.x * blockDim.x + threadIdx.x;
  int edge = gid >> 5;
  int lane = threadIdx.x & 31;
  if (edge >= EE) return;
  int sN = src[edge], dN = dst[edge];
  float a0 = ebuf[2 * edge]     / sbuf[2 * dN];
  float a1 = ebuf[2 * edge + 1] / sbuf[2 * dN + 1];
  int c = lane * 4;
  float a = (c < 64) ? a0 : a1;
  const float4 hv = *(const float4*)(h + (size_t)sN * 128 + c);
  float* ap = agg + (size_t)dN * 128 + c;
  atomicAdd(ap + 0, hv.x * a);
  atomicAdd(ap + 1, hv.y * a);
  atomicAdd(ap + 2, hv.z * a);
  atomicAdd(ap + 3, hv.w * a);
}

// ---------------------------------------------------------------------------
// Epilogue (layers 1-2): in-place x = ELU(BN(agg + b)).
// ---------------------------------------------------------------------------
__global__ void bn_elu(float* __restrict__ agg, const float* __restrict__ b,
                       const float* __restrict__ g, const float* __restrict__ be,
                       const float* __restrict__ mn, const float* __restrict__ vr) {
  int t = blockIdx.x * blockDim.x + threadIdx.x;
  if (t >= NN * 128) return;
  int c = t & 127;
  float x = agg[t] + b[c];
  x = (x - mn[c]) * rsqrtf(vr[c] + 1e-5f) * g[c] + be[c];
  agg[t] = x > 0.f ? x : expm1f(x);
}

// ---------------------------------------------------------------------------
// Final epilogue: out[n,f] = mean over heads of (agg + b3).
// ---------------------------------------------------------------------------
__global__ void final_mean(const float* __restrict__ agg, const float* __restrict__ b3,
                           float* __restrict__ out) {
  int t = blockIdx.x * blockDim.x + threadIdx.x;
  if (t >= NN * 64) return;
  int n = t >> 6, f = t & 63;
  float v0 = agg[(size_t)n * 128 + f]      + b3[f];
  float v1 = agg[(size_t)n * 128 + 64 + f] + b3[64 + f];
  out[t] = 0.5f * (v0 + v1);
}

// ---------------------------------------------------------------------------
extern "C" void kernel_launch(void* const* d_in, const int* in_sizes, int n_in,
                              void* d_out, int out_size, void* d_ws, size_t ws_size,
                              hipStream_t stream) {
  const float* x    = (const float*)d_in[0];
  const int*   src  = (const int*)d_in[1];
  const int*   dst  = (const int*)d_in[2];
  const float* W1   = (const float*)d_in[3];
  const float* al1  = (const float*)d_in[4];
  const float* ar1  = (const float*)d_in[5];
  const float* b1   = (const float*)d_in[6];
  const float* g1   = (const float*)d_in[7];
  const float* be1  = (const float*)d_in[8];
  const float* m1   = (const float*)d_in[9];
  const float* v1   = (const float*)d_in[10];
  const float* W2   = (const float*)d_in[11];
  const float* al2  = (const float*)d_in[12];
  const float* ar2  = (const float*)d_in[13];
  const float* b2   = (const float*)d_in[14];
  const float* g2   = (const float*)d_in[15];
  const float* be2  = (const float*)d_in[16];
  const float* m2   = (const float*)d_in[17];
  const float* v2   = (const float*)d_in[18];
  const float* W3   = (const float*)d_in[19];
  const float* al3  = (const float*)d_in[20];
  const float* ar3  = (const float*)d_in[21];
  const float* b3   = (const float*)d_in[22];
  float* out = (float*)d_out;

  float* ws   = (float*)d_ws;
  float* h    = ws;                        // NN*128
  float* agg  = h + (size_t)NN * 128;      // NN*128
  float* ebuf = agg + (size_t)NN * 128;    // EE*2
  float* el   = ebuf + (size_t)EE * 2;     // NN*2
  float* er   = el + (size_t)NN * 2;       // NN*2
  float* mbuf = er + (size_t)NN * 2;       // NN*2
  float* sbuf = mbuf + (size_t)NN * 2;     // NN*2

  const int waves = NN / 16;                       // 3125 (exact)
  dim3 gemm_grid((waves * 32 + 127) / 128), gemm_blk(128);
  dim3 node2_grid((NN * 2 + 255) / 256);
  dim3 nodeF_grid((NN * 128 + 255) / 256);
  dim3 edge_grid((EE + 255) / 256);
  dim3 edgeW_grid(((size_t)EE * 32 + 255) / 256);
  dim3 blk(256);

  auto run_layer = [&](const float* X, int K, const float* W,
                       const float* al, const float* ar) {
    gemm_wmma_f32<<<gemm_grid, gemm_blk, 0, stream>>>(X, W, h, K);
    eler_kernel<<<node2_grid, blk, 0, stream>>>(h, al, ar, el, er);
    init_kernel<<<nodeF_grid, blk, 0, stream>>>(agg, mbuf, sbuf);
    edge_pass1<<<edge_grid, blk, 0, stream>>>(src, dst, el, er, ebuf, mbuf);
    edge_pass2<<<edge_grid, blk, 0, stream>>>(dst, ebuf, mbuf, sbuf);
    edge_pass3<<<edgeW_grid, blk, 0, stream>>>(src, dst, ebuf, sbuf, h, agg);
  };

  // Layer 1: input x [N,256]
  run_layer(x, 256, W1, al1, ar1);
  bn_elu<<<nodeF_grid, blk, 0, stream>>>(agg, b1, g1, be1, m1, v1);

  // Layer 2: input agg (in-place epilogue output) [N,128]
  run_layer(agg, 128, W2, al2, ar2);
  bn_elu<<<nodeF_grid, blk, 0, stream>>>(agg, b2, g2, be2, m2, v2);

  // Layer 3
  run_layer(agg, 128, W3, al3, ar3);
  final_mean<<<(dim3)((NN * 64 + 255) / 256), blk, 0, stream>>>(agg, b3, out);
}